// SparseAttentionLayer_54013508714716
// MI455X (gfx1250) — compile-verified
//
#include <hip/hip_runtime.h>
#include <math.h>

#define NN      50000
#define HID     64
#define HEADS   4
#define HDIM    16

typedef __attribute__((ext_vector_type(2))) float v2f;
typedef __attribute__((ext_vector_type(8))) float v8f;

__device__ __forceinline__ v8f wmma4(v2f a, v2f b, v8f c) {
    // V_WMMA_F32_16X16X4_F32 : D = A(16x4) * B(4x16) + C(16x16), all fp32
    return __builtin_amdgcn_wmma_f32_16x16x4_f32(
        false, a, false, b, (short)0, c, false, false);
}

// ---------------------------------------------------------------------------
// Kernel 1: Q/K/V = x @ W{q,k,v}  and per-node gate = sigmoid(relu(x@Wg1+b1)@Wg2+b2)
// block = 128 threads (4 waves), each block does 16 rows; wave w does cols [16w,16w+16)
// Requires N % 16 == 0 (50000 = 3125*16) so EXEC is all-ones for WMMA.
// Gate accumulation is hoisted into ONE wave-0-only region after the main loop
// (single saveexec instead of one per K-step) so the hot loop is a pure
// 3-WMMA chain the scheduler can clause.
// ---------------------------------------------------------------------------
__global__ __launch_bounds__(128)
void qkv_gate_kernel(const float* __restrict__ x,
                     const float* __restrict__ Wq, const float* __restrict__ Wk,
                     const float* __restrict__ Wv,
                     const float* __restrict__ Wg1, const float* __restrict__ bg1,
                     const float* __restrict__ Wg2, const float* __restrict__ bg2,
                     float* __restrict__ Q, float* __restrict__ K,
                     float* __restrict__ V, float* __restrict__ gate) {
    __shared__ float xs[16 * 68];               // stride 68 -> conflict-free A reads
    const int row0 = blockIdx.x * 16;
    const int tid  = threadIdx.x;

    // stage 16x64 x-tile into LDS (float4)
    for (int i = tid; i < 16 * 16; i += 128) {
        const int r = i >> 4, c4 = i & 15;
        const float4 v = ((const float4*)(x + (size_t)(row0 + r) * HID))[c4];
        float* dst = &xs[r * 68 + c4 * 4];
        dst[0] = v.x; dst[1] = v.y; dst[2] = v.z; dst[3] = v.w;
    }
    __syncthreads();

    const int wave = tid >> 5;
    const int lane = tid & 31;
    const int m    = lane & 15;      // A row
    const int kh   = lane >> 4;      // K half select
    const int n    = lane & 15;      // B/D column
    const int n0   = wave * 16;

    v8f accQ = {}, accK = {}, accV = {};
    #pragma unroll
    for (int kk = 0; kk < 16; ++kk) {
        const int k0 = kk * 4 + kh * 2;
        v2f a;
        a.x = xs[m * 68 + k0];
        a.y = xs[m * 68 + k0 + 1];
        v2f bq, bk, bv;
        bq.x = Wq[(size_t)(k0    ) * HID + n0 + n];
        bq.y = Wq[(size_t)(k0 + 1) * HID + n0 + n];
        bk.x = Wk[(size_t)(k0    ) * HID + n0 + n];
        bk.y = Wk[(size_t)(k0 + 1) * HID + n0 + n];
        bv.x = Wv[(size_t)(k0    ) * HID + n0 + n];
        bv.y = Wv[(size_t)(k0 + 1) * HID + n0 + n];
        accQ = wmma4(a, bq, accQ);
        accK = wmma4(a, bk, accK);
        accV = wmma4(a, bv, accV);
    }

    #pragma unroll
    for (int v = 0; v < 8; ++v) {
        const int    r = row0 + v + kh * 8;
        const size_t o = (size_t)r * HID + n0 + n;
        Q[o] = accQ[v]; K[o] = accK[v]; V[o] = accV[v];
    }

    // ---- gate MLP: single divergent region, wave-uniform predicate ----
    if (wave == 0) {
        v8f accG = {};
        #pragma unroll
        for (int kk = 0; kk < 16; ++kk) {
            const int k0 = kk * 4 + kh * 2;
            v2f a;
            a.x = xs[m * 68 + k0];
            a.y = xs[m * 68 + k0 + 1];
            v2f bg;
            bg.x = Wg1[(size_t)(k0    ) * 16 + n];
            bg.y = Wg1[(size_t)(k0 + 1) * 16 + n];
            accG = wmma4(a, bg, accG);           // x @ Wg1 (64x16)
        }
        const float w2 = Wg2[n];
        const float b1 = bg1[n];
        const float b2 = bg2[0];
        #pragma unroll
        for (int v = 0; v < 8; ++v) {
            float g = fmaxf(accG[v] + b1, 0.0f) * w2;   // relu then * Wg2[n]
            g += __shfl_xor(g, 1, 32);
            g += __shfl_xor(g, 2, 32);
            g += __shfl_xor(g, 4, 32);
            g += __shfl_xor(g, 8, 32);                  // sum over n=0..15
            if (n == 0) {
                const int r = row0 + v + kh * 8;
                gate[r] = 1.0f / (1.0f + __expf(-(g + b2)));
            }
        }
    }
}

// ---------------------------------------------------------------------------
// Kernel 2: per-(edge,head) attention score + segment sum over destination col
// ---------------------------------------------------------------------------
__global__ void score_kernel(const float* __restrict__ Q, const float* __restrict__ K,
                             const float* __restrict__ gate, const int* __restrict__ ei,
                             float* __restrict__ scores, float* __restrict__ seg, int E) {
    const int t = blockIdx.x * blockDim.x + threadIdx.x;
    if (t >= E * HEADS) return;
    const int e = t >> 2, h = t & 3;
    const int row = ei[e], col = ei[E + e];
    const float4* q = (const float4*)(Q + (size_t)row * HID + h * HDIM);
    const float4* k = (const float4*)(K + (size_t)col * HID + h * HDIM);
    float s = 0.0f;
    #pragma unroll
    for (int i = 0; i < 4; ++i) {
        const float4 a = q[i], b = k[i];
        s += a.x * b.x + a.y * b.y + a.z * b.z + a.w * b.w;
    }
    s = s * 0.25f * gate[row];                  // 1/sqrt(16) * gate
    scores[t] = s;
    atomicAdd(&seg[col * HEADS + h], s);
}

// ---------------------------------------------------------------------------
// Kernel 3: exp(score - segsum[col]) and denominator accumulation
// ---------------------------------------------------------------------------
__global__ void expnorm_kernel(const int* __restrict__ ei, float* __restrict__ scores,
                               const float* __restrict__ seg, float* __restrict__ denom,
                               int E) {
    const int t = blockIdx.x * blockDim.x + threadIdx.x;
    if (t >= E * HEADS) return;
    const int e = t >> 2, h = t & 3;
    const int col = ei[E + e];
    const float es = __expf(scores[t] - seg[col * HEADS + h]);
    scores[t] = es;                              // overwrite in place with exp_s
    atomicAdd(&denom[col * HEADS + h], es);
}

// ---------------------------------------------------------------------------
// Kernel 4: agg[col] += (exp_s / (denom[col]+eps)) * V[col]
// ---------------------------------------------------------------------------
__global__ void agg_kernel(const int* __restrict__ ei, const float* __restrict__ scores,
                           const float* __restrict__ denom, const float* __restrict__ V,
                           float* __restrict__ agg, int E) {
    const int t = blockIdx.x * blockDim.x + threadIdx.x;
    if (t >= E * HEADS) return;
    const int e = t >> 2, h = t & 3;
    const int col = ei[E + e];
    const float w = scores[t] / (denom[col * HEADS + h] + 1e-8f);
    const float* v = V   + (size_t)col * HID + h * HDIM;
    float*       a = agg + (size_t)col * HID + h * HDIM;
    #pragma unroll
    for (int i = 0; i < HDIM; ++i) atomicAdd(a + i, w * v[i]);
}

// ---------------------------------------------------------------------------
// Kernel 5: out = agg @ Wo + bo + x   (WMMA fp32, same tiling as kernel 1)
// ---------------------------------------------------------------------------
__global__ __launch_bounds__(128)
void out_kernel(const float* __restrict__ agg, const float* __restrict__ Wo,
                const float* __restrict__ bo, const float* __restrict__ x,
                float* __restrict__ out) {
    __shared__ float as_[16 * 68];
    const int row0 = blockIdx.x * 16;
    const int tid  = threadIdx.x;

    for (int i = tid; i < 16 * 16; i += 128) {
        const int r = i >> 4, c4 = i & 15;
        const float4 v = ((const float4*)(agg + (size_t)(row0 + r) * HID))[c4];
        float* dst = &as_[r * 68 + c4 * 4];
        dst[0] = v.x; dst[1] = v.y; dst[2] = v.z; dst[3] = v.w;
    }
    __syncthreads();

    const int wave = tid >> 5;
    const int lane = tid & 31;
    const int m    = lane & 15;
    const int kh   = lane >> 4;
    const int n    = lane & 15;
    const int n0   = wave * 16;

    v8f acc = {};
    #pragma unroll
    for (int kk = 0; kk < 16; ++kk) {
        const int k0 = kk * 4 + kh * 2;
        v2f a;
        a.x = as_[m * 68 + k0];
        a.y = as_[m * 68 + k0 + 1];
        v2f b;
        b.x = Wo[(size_t)(k0    ) * HID + n0 + n];
        b.y = Wo[(size_t)(k0 + 1) * HID + n0 + n];
        acc = wmma4(a, b, acc);
    }

    const float bias = bo[n0 + n];
    #pragma unroll
    for (int v = 0; v < 8; ++v) {
        const int    r = row0 + v + kh * 8;
        const size_t o = (size_t)r * HID + n0 + n;
        out[o] = acc[v] + bias + x[o];          // residual
    }
}

__global__ void zero_kernel(float* __restrict__ p, int nfloats) {
    const int i = blockIdx.x * blockDim.x + threadIdx.x;
    if (i < nfloats) p[i] = 0.0f;
}

// ---------------------------------------------------------------------------
extern "C" void kernel_launch(void* const* d_in, const int* in_sizes, int n_in,
                              void* d_out, int out_size, void* d_ws, size_t ws_size,
                              hipStream_t stream) {
    const float* x   = (const float*)d_in[0];
    const int*   ei  = (const int*)  d_in[1];
    const float* Wq  = (const float*)d_in[2];
    const float* Wk  = (const float*)d_in[3];
    const float* Wv  = (const float*)d_in[4];
    const float* Wo  = (const float*)d_in[5];
    const float* bo  = (const float*)d_in[6];
    const float* Wg1 = (const float*)d_in[7];
    const float* bg1 = (const float*)d_in[8];
    const float* Wg2 = (const float*)d_in[9];
    const float* bg2 = (const float*)d_in[10];

    const int N = in_sizes[0] / HID;     // 50000 (multiple of 16)
    const int E = in_sizes[1] / 2;       // 800000

    // workspace layout (floats); seg/denom/agg kept contiguous for one zero pass
    float* ws     = (float*)d_ws;
    float* Q      = ws;                          // N*64
    float* K      = Q      + (size_t)N * HID;    // N*64
    float* V      = K      + (size_t)N * HID;    // N*64
    float* gate   = V      + (size_t)N * HID;    // N
    float* scores = gate   + (size_t)N;          // E*4
    float* seg    = scores + (size_t)E * HEADS;  // N*4  -+
    float* denom  = seg    + (size_t)N * HEADS;  // N*4   | contiguous zero region
    float* agg    = denom  + (size_t)N * HEADS;  // N*64 -+

    const int zcount = N * (HEADS + HEADS + HID);
    zero_kernel<<<(zcount + 255) / 256, 256, 0, stream>>>(seg, zcount);

    qkv_gate_kernel<<<N / 16, 128, 0, stream>>>(x, Wq, Wk, Wv, Wg1, bg1, Wg2, bg2,
                                                Q, K, V, gate);

    const int eh = E * HEADS;
    score_kernel  <<<(eh + 255) / 256, 256, 0, stream>>>(Q, K, gate, ei, scores, seg, E);
    expnorm_kernel<<<(eh + 255) / 256, 256, 0, stream>>>(ei, scores, seg, denom, E);
    agg_kernel    <<<(eh + 255) / 256, 256, 0, stream>>>(ei, scores, denom, V, agg, E);

    out_kernel<<<N / 16, 128, 0, stream>>>(agg, Wo, bo, x, (float*)d_out);
}